// Kernel_65481071398031
// MI455X (gfx1250) — compile-verified
//
#include <hip/hip_runtime.h>
#include <math.h>

// ---------------------------------------------------------------------------
// 20-qubit state-vector simulator for the PennyLane-style circuit, optimized
// for MI455X (gfx1250, wave32):
//  - 4-wire gate groups fused into 16x16 complex unitaries, applied with
//    V_WMMA_F32_16X16X4_F32 (16 WMMAs per 16x16 column tile, full f32).
//  - Each layer's CNOT ring composes into one GF(2)-linear bit permutation,
//    folded into the next layer's first group-pass load addressing.
//  - Final ring permutation skipped: |<s1|s2>| invariant under common bijection.
//  - Everything stays L2-resident (32MB working set << 192MB L2).
//  - P / PERM are template params: constant-folded addressing, no in-loop
//    branches; P==0 specialization uses b64 loads + b128 stores.
// ---------------------------------------------------------------------------

typedef float v2f __attribute__((ext_vector_type(2)));
typedef float v4f __attribute__((ext_vector_type(4)));
typedef float v8f __attribute__((ext_vector_type(8)));

struct c32 { float re, im; };
__device__ __forceinline__ c32 cmul(c32 a, c32 b){
  return { fmaf(a.re, b.re, -a.im * b.im), fmaf(a.re, b.im, a.im * b.re) };
}
__device__ __forceinline__ c32 cadd(c32 a, c32 b){ return { a.re + b.re, a.im + b.im }; }

// Rot(phi,theta,omega) = RZ(omega) @ RY(theta) @ RZ(phi)
__device__ __forceinline__ void rotMat(float phi, float theta, float omega, c32 M[2][2]){
  float ct = cosf(0.5f * theta), st = sinf(0.5f * theta);
  float apo = 0.5f * (phi + omega);
  float pmo = 0.5f * (phi - omega);
  float ca = cosf(apo), sa = sinf(apo);
  float cp = cosf(pmo), sp = sinf(pmo);
  M[0][0] = {  ct * ca, -ct * sa };   //  ct * e^{-i(phi+omega)/2}
  M[0][1] = { -st * cp, -st * sp };   // -st * e^{+i(phi-omega)/2}
  M[1][0] = {  st * cp, -st * sp };   //  st * e^{-i(phi-omega)/2}
  M[1][1] = {  ct * ca,  ct * sa };   //  ct * e^{+i(phi+omega)/2}
}

// Encoding E = RY(0.5 v) @ RX(v)
__device__ __forceinline__ void encMat(float v, c32 E[2][2]){
  float c2 = cosf(0.5f * v),  s2 = sinf(0.5f * v);    // RX(v)
  float c3 = cosf(0.25f * v), s3 = sinf(0.25f * v);   // RY(0.5 v)
  E[0][0] = {  c3 * c2,  s3 * s2 };
  E[0][1] = { -s3 * c2, -c3 * s2 };
  E[1][0] = {  s3 * c2, -c3 * s2 };
  E[1][1] = {  c3 * c2, -s3 * s2 };
}

__device__ __forceinline__ void mm2(const c32 A[2][2], const c32 B[2][2], c32 C[2][2]){
#pragma unroll
  for (int i = 0; i < 2; ++i)
#pragma unroll
    for (int j = 0; j < 2; ++j)
      C[i][j] = cadd(cmul(A[i][0], B[0][j]), cmul(A[i][1], B[1][j]));
}

// Inverse of the full CNOT ring permutation (wire w <-> flat bit 19-w):
// ring: for w=0..18: b_{w+1}^=b_w; then b_0^=b_19.
__device__ __forceinline__ unsigned permInv(unsigned i){
  unsigned j = i ^ ((i >> 1) & 0x7FFFFu);
  if (i & 1u) j ^= 0xC0000u;  // bits 18,19 get XOR of bit 0
  return j;
}

// ---------------------------------------------------------------------------
// Build the 40 fused 16x16 group unitaries: [state(2)][layer(4)][group(5)].
// Layer 0 folds the RX/RY encoding into the Rot gates.  No CNOTs here: the
// whole ring is handled by the permutation folding.
// ---------------------------------------------------------------------------
__global__ __launch_bounds__(256) void qc_build_unitaries(
    const float* __restrict__ x, const float* __restrict__ y,
    const float* __restrict__ prm, float* __restrict__ UR, float* __restrict__ UI)
{
  unsigned uid   = blockIdx.x;        // 0..39
  unsigned state = uid / 20u;
  unsigned rem   = uid % 20u;
  unsigned layer = rem / 5u;
  unsigned group = rem % 5u;
  const float* v = state ? y : x;

  unsigned e = threadIdx.x;           // one matrix entry per thread
  unsigned q = e >> 4, r = e & 15u;   // U[q][r]
  c32 acc = { 1.f, 0.f };
#pragma unroll
  for (int k = 0; k < 4; ++k){
    unsigned w = 4u * group + (unsigned)k;        // wire; group bit 3-k
    const float* pw = prm + (layer * 20u + w) * 3u;
    c32 M[2][2];
    rotMat(pw[0], pw[1], pw[2], M);
    if (layer == 0u){
      c32 E[2][2]; encMat(v[w], E);
      c32 P2[2][2]; mm2(M, E, P2);                // Rot * (RY*RX)
      M[0][0] = P2[0][0]; M[0][1] = P2[0][1];
      M[1][0] = P2[1][0]; M[1][1] = P2[1][1];
    }
    unsigned bq = (q >> (3 - k)) & 1u;
    unsigned br = (r >> (3 - k)) & 1u;
    acc = cmul(acc, M[bq][br]);
  }
  UR[(uid << 8) + e] = acc.re;
  UI[(uid << 8) + e] = acc.im;
}

// ---------------------------------------------------------------------------
// Initialize both states in buffer 0 to |0...0>.
// ---------------------------------------------------------------------------
__global__ __launch_bounds__(256) void qc_init_state(float* b0R, float* b0I){
  unsigned tid = blockIdx.x * 256u + threadIdx.x;       // covers 2<<20
  b0R[tid] = ((tid & 0xFFFFFu) == 0u) ? 1.f : 0.f;
  b0I[tid] = 0.f;
}

// ---------------------------------------------------------------------------
// One group pass: out[i(r',c)] = sum_r U[r'][r] * src[maybePerm(i(r,c))]
// Group acts on flat bits [P..P+3].  One wave = 16x16 column tiles.
// Complex matmul via 4 real WMMA accumulations per K-chunk of 4.
// ---------------------------------------------------------------------------
template<int P, bool PERM>
__global__ __launch_bounds__(256) void qc_gate_pass(
    const float* srcR, const float* srcI, float* dstR, float* dstI,
    const float* __restrict__ UR, const float* __restrict__ UI, int lg)
{
  const unsigned TPW = 4;                       // tiles per wave (A reuse)
  unsigned wave   = threadIdx.x >> 5;
  unsigned lane   = threadIdx.x & 31u;
  unsigned laneHi = lane >> 4;                  // 0: K+{0,1} / rows 0-7; 1: K+{2,3} / rows 8-15
  unsigned nl     = lane & 15u;                 // column within tile / A row
  unsigned stateOff = blockIdx.y << 20;

  const float* sR = srcR + stateOff;
  const float* sI = srcI + stateOff;
  float* dR = dstR + stateOff;
  float* dI = dstI + stateOff;
  unsigned uoff = (blockIdx.y * 20u + (unsigned)lg) << 8;
  const float* uR = UR + uoff;
  const float* uI = UI + uoff;

  // A operand (U columns 4s..4s+3), ISA 16x4 f32 layout.
  v2f aR[4], aI[4], aIn[4];
#pragma unroll
  for (int s = 0; s < 4; ++s){
    unsigned k = 4u * s + 2u * laneHi;
    unsigned o = nl * 16u + k;                  // row-major U, 8B aligned (k even)
    aR[s] = *(const v2f*)(uR + o);
    aI[s] = *(const v2f*)(uI + o);
    aIn[s].x = -aI[s].x; aIn[s].y = -aI[s].y;   // f32 WMMA has no A-neg modifier
  }

  constexpr unsigned loMask = (1u << P) - 1u;
  unsigned tileBase = (blockIdx.x * 8u + wave) * TPW;
#pragma unroll 1
  for (unsigned it = 0; it < TPW; ++it){
    unsigned t = tileBase + it;
    unsigned c = t * 16u + nl;                                   // global column
    unsigned colBase = ((c >> P) << (P + 4)) | (c & loMask);     // i = colBase | (r<<P)
    v8f cR = {}; v8f cI = {};
#pragma unroll
    for (int s = 0; s < 4; ++s){
      unsigned k  = 4u * s + 2u * laneHi;
      v2f bR, bI;
      if constexpr (P == 0 && !PERM){
        // rows k, k+1 are adjacent floats -> single b64 load per plane
        unsigned i0 = colBase | k;
        bR = *(const v2f*)(sR + i0);
        bI = *(const v2f*)(sI + i0);
      } else {
        unsigned i0 = colBase | (k << P);
        unsigned i1 = colBase | ((k + 1u) << P);
        if constexpr (PERM){ i0 = permInv(i0); i1 = permInv(i1); }
        bR.x = sR[i0]; bR.y = sR[i1];
        bI.x = sI[i0]; bI.y = sI[i1];
      }
      // Dr += Ur*Br - Ui*Bi ; Di += Ur*Bi + Ui*Br
      cR = __builtin_amdgcn_wmma_f32_16x16x4_f32(false, aR[s],  false, bR, (short)0, cR, false, false);
      cR = __builtin_amdgcn_wmma_f32_16x16x4_f32(false, aIn[s], false, bI, (short)0, cR, false, false);
      cI = __builtin_amdgcn_wmma_f32_16x16x4_f32(false, aR[s],  false, bI, (short)0, cI, false, false);
      cI = __builtin_amdgcn_wmma_f32_16x16x4_f32(false, aI[s],  false, bR, (short)0, cI, false, false);
    }
    if constexpr (P == 0){
      // rows m = 8*laneHi + 0..7 are 8 consecutive floats -> two b128 stores
      unsigned o = colBase | (8u * laneHi);
      v4f lo, hi;
      lo.x = cR[0]; lo.y = cR[1]; lo.z = cR[2]; lo.w = cR[3];
      hi.x = cR[4]; hi.y = cR[5]; hi.z = cR[6]; hi.w = cR[7];
      *(v4f*)(dR + o)      = lo;
      *(v4f*)(dR + o + 4u) = hi;
      lo.x = cI[0]; lo.y = cI[1]; lo.z = cI[2]; lo.w = cI[3];
      hi.x = cI[4]; hi.y = cI[5]; hi.z = cI[6]; hi.w = cI[7];
      *(v4f*)(dI + o)      = lo;
      *(v4f*)(dI + o + 4u) = hi;
    } else {
#pragma unroll
      for (int v = 0; v < 8; ++v){
        unsigned m = (unsigned)v + 8u * laneHi;   // C/D row layout
        unsigned o = colBase | (m << P);
        dR[o] = cR[v];
        dI[o] = cI[v];
      }
    }
  }
}

// ---------------------------------------------------------------------------
// Deterministic two-stage reduction of <s1|s2> and |.|
// ---------------------------------------------------------------------------
__global__ __launch_bounds__(256) void qc_reduce_partial(
    const float* __restrict__ bR, const float* __restrict__ bI, float* partials)
{
  const unsigned N = 1u << 20;
  float accRe = 0.f, accIm = 0.f;
  for (unsigned i = blockIdx.x * 256u + threadIdx.x; i < N; i += 256u * 256u){
    float r1 = bR[i],     i1 = bI[i];
    float r2 = bR[i + N], i2 = bI[i + N];
    accRe = fmaf(r1, r2, fmaf(i1, i2, accRe));      // Re(conj(s1)*s2)
    accIm = fmaf(r1, i2, fmaf(-i1, r2, accIm));     // Im(conj(s1)*s2)
  }
  __shared__ float shR[256], shI[256];
  shR[threadIdx.x] = accRe; shI[threadIdx.x] = accIm;
  __syncthreads();
  for (int s = 128; s > 0; s >>= 1){
    if ((int)threadIdx.x < s){
      shR[threadIdx.x] += shR[threadIdx.x + s];
      shI[threadIdx.x] += shI[threadIdx.x + s];
    }
    __syncthreads();
  }
  if (threadIdx.x == 0){
    partials[blockIdx.x]        = shR[0];
    partials[256u + blockIdx.x] = shI[0];
  }
}

__global__ __launch_bounds__(256) void qc_reduce_final(const float* partials, float* out){
  __shared__ float shR[256], shI[256];
  shR[threadIdx.x] = partials[threadIdx.x];
  shI[threadIdx.x] = partials[256u + threadIdx.x];
  __syncthreads();
  for (int s = 128; s > 0; s >>= 1){
    if ((int)threadIdx.x < s){
      shR[threadIdx.x] += shR[threadIdx.x + s];
      shI[threadIdx.x] += shI[threadIdx.x + s];
    }
    __syncthreads();
  }
  if (threadIdx.x == 0) out[0] = sqrtf(shR[0] * shR[0] + shI[0] * shI[0]);
}

// ---------------------------------------------------------------------------
static inline void launch_gate(const float* sR, const float* sI, float* dR, float* dI,
                               const float* UR, const float* UI, int lg, int g,
                               bool perm, hipStream_t stream)
{
  dim3 grid(128, 2);  // 128 blocks x 8 waves x 4 tiles = 4096 tiles; y = state
  switch (g){
    case 0:
      if (perm) qc_gate_pass<16, true ><<<grid, 256, 0, stream>>>(sR, sI, dR, dI, UR, UI, lg);
      else      qc_gate_pass<16, false><<<grid, 256, 0, stream>>>(sR, sI, dR, dI, UR, UI, lg);
      break;
    case 1: qc_gate_pass<12, false><<<grid, 256, 0, stream>>>(sR, sI, dR, dI, UR, UI, lg); break;
    case 2: qc_gate_pass< 8, false><<<grid, 256, 0, stream>>>(sR, sI, dR, dI, UR, UI, lg); break;
    case 3: qc_gate_pass< 4, false><<<grid, 256, 0, stream>>>(sR, sI, dR, dI, UR, UI, lg); break;
    default: qc_gate_pass<0, false><<<grid, 256, 0, stream>>>(sR, sI, dR, dI, UR, UI, lg); break;
  }
}

extern "C" void kernel_launch(void* const* d_in, const int* in_sizes, int n_in,
                              void* d_out, int out_size, void* d_ws, size_t ws_size,
                              hipStream_t stream) {
  (void)in_sizes; (void)n_in; (void)out_size; (void)ws_size;
  const float* x   = (const float*)d_in[0];   // (20,)
  const float* yv  = (const float*)d_in[1];   // (20,)
  const float* prm = (const float*)d_in[2];   // (4,20,3)
  float* out = (float*)d_out;

  // Workspace layout (floats): 2 ping-pong buffers x 2 states x (re,im planes)
  float* ws  = (float*)d_ws;
  const size_t M = (size_t)1 << 20;
  float* b0R = ws;                 float* b0I = ws + 2 * M;
  float* b1R = ws + 4 * M;         float* b1I = ws + 6 * M;
  float* UR  = ws + 8 * M;         float* UI  = UR + 40 * 256;
  float* partials = UI + 40 * 256; // 512 floats

  qc_build_unitaries<<<40, 256, 0, stream>>>(x, yv, prm, UR, UI);
  qc_init_state<<<8192, 256, 0, stream>>>(b0R, b0I);

  // Layer 0 (encoding fused): in-place on buf0, no perm.
  for (int g = 0; g < 5; ++g)
    launch_gate(b0R, b0I, b0R, b0I, UR, UI, 0 + g, g, false, stream);
  // Layer 1: ring perm of layer 0 folded into first pass (buf0 -> buf1).
  launch_gate(b0R, b0I, b1R, b1I, UR, UI, 5, 0, true, stream);
  for (int g = 1; g < 5; ++g)
    launch_gate(b1R, b1I, b1R, b1I, UR, UI, 5 + g, g, false, stream);
  // Layer 2: perm folded (buf1 -> buf0).
  launch_gate(b1R, b1I, b0R, b0I, UR, UI, 10, 0, true, stream);
  for (int g = 1; g < 5; ++g)
    launch_gate(b0R, b0I, b0R, b0I, UR, UI, 10 + g, g, false, stream);
  // Layer 3: perm folded (buf0 -> buf1).  Final ring perm after layer 3 is a
  // common bijection of both states -> overlap invariant -> skipped.
  launch_gate(b0R, b0I, b1R, b1I, UR, UI, 15, 0, true, stream);
  for (int g = 1; g < 5; ++g)
    launch_gate(b1R, b1I, b1R, b1I, UR, UI, 15 + g, g, false, stream);

  qc_reduce_partial<<<256, 256, 0, stream>>>(b1R, b1I, partials);
  qc_reduce_final<<<1, 256, 0, stream>>>(partials, out);
}